// AspectNeighborAttention_64493228917346
// MI455X (gfx1250) — compile-verified
//
#include <hip/hip_runtime.h>
#include <hip/hip_bf16.h>

// ---------------------------------------------------------------------------
// AspectNeighborAttention for MI455X (gfx1250, wave32, WMMA)
// B=32, L=128, H=768, E=64
// ---------------------------------------------------------------------------

typedef float v2f __attribute__((ext_vector_type(2)));
typedef float v8f __attribute__((ext_vector_type(8)));

#define BB 32
#define LL 128
#define HH 768
#define EE 64
#define MM (BB * LL)      // 4096 rows
#define NT (HH / 16)      // 48 n-tiles
#define MT (MM / 16)      // 256 m-tiles

static __device__ __forceinline__ v8f wmma4(v2f a, v2f b, v8f c) {
  // D = A(16x4,f32) x B(4x16,f32) + C(16x16,f32)
  return __builtin_amdgcn_wmma_f32_16x16x4_f32(
      /*neg_a=*/false, a, /*neg_b=*/false, b,
      /*c_mod=*/(short)0, c, /*reuse_a=*/false, /*reuse_b=*/false);
}

// ---------------------------------------------------------------------------
// K0: out <- bert  (rows not updated keep bert)
// ---------------------------------------------------------------------------
__global__ __launch_bounds__(256) void copy_f4_kernel(const float4* __restrict__ src,
                                                      float4* __restrict__ dst, int n) {
  for (int t = blockIdx.x * blockDim.x + threadIdx.x; t < n; t += gridDim.x * blockDim.x)
    dst[t] = src[t];
}

// ---------------------------------------------------------------------------
// K1 / K3: generic GEMM  out[m, n] = sum_k in[src(m), k] * W[n*ldw + k] (+ bias[n])
// src(m) optionally applies roll(-1) within each batch of L rows.
// One wave per 16x16 output tile; K stepped by 4 with v_wmma_f32_16x16x4_f32.
// ---------------------------------------------------------------------------
__global__ __launch_bounds__(256) void gemm_wmma_kernel(const float* __restrict__ in,
                                                        int rollShift,
                                                        const float* __restrict__ W, int ldw,
                                                        const float* __restrict__ bias,
                                                        float* __restrict__ out, int K) {
  const int lane = threadIdx.x & 31;
  const int wave = threadIdx.x >> 5;
  const int tile = blockIdx.x * 8 + wave;       // MT*NT = 12288 tiles, 1536 blocks
  const int mt = tile / NT;
  const int nt = tile % NT;
  const int half = lane >> 4;
  const int col = lane & 15;

  int mrow = mt * 16 + col;                     // A-matrix row held by this lane
  if (rollShift) {
    int b = mrow >> 7, i = mrow & (LL - 1);
    mrow = (b << 7) | ((i + 1) & (LL - 1));     // roll(z, -1, axis=1)
  }
  const float* arow = in + (size_t)mrow * K;
  const float* brow = W + (size_t)(nt * 16 + col) * ldw;
  const int koff = 2 * half;                    // k = v + 2*(lane/16)

  v8f acc = {};
#pragma unroll 4
  for (int k = 0; k < K; k += 4) {
    if ((k & 63) == 0 && k + 64 < K) {
      __builtin_prefetch(arow + k + 64, 0, 0);
      __builtin_prefetch(brow + k + 64, 0, 0);
    }
    v2f a = *(const v2f*)(arow + k + koff);
    v2f b = *(const v2f*)(brow + k + koff);
    acc = wmma4(a, b, acc);
  }

  const float bv = bias ? bias[nt * 16 + col] : 0.0f;
  const int n = nt * 16 + col;
#pragma unroll
  for (int r = 0; r < 8; ++r) {
    int m = mt * 16 + r + 8 * half;             // C/D layout: M = r + 8*(lane/16)
    out[(size_t)m * HH + n] = acc[r] + bv;
  }
}

// ---------------------------------------------------------------------------
// K2a: s_i[row] = zs[row,:].wa[0:H],  s_j[row] = zs[row,:].wa[H:2H]
// wave per row, shuffle reduction (wave32)
// ---------------------------------------------------------------------------
__global__ __launch_bounds__(256) void score_ij_kernel(const float* __restrict__ zs,
                                                       const float* __restrict__ wa,
                                                       float* __restrict__ s_i,
                                                       float* __restrict__ s_j) {
  const int lane = threadIdx.x & 31;
  const int wave = threadIdx.x >> 5;
  const int row = blockIdx.x * 8 + wave;        // 512 blocks -> 4096 rows
  const float* zrow = zs + (size_t)row * HH;
  float ai = 0.0f, aj = 0.0f;
#pragma unroll
  for (int t = lane; t < HH; t += 32) {
    float zv = zrow[t];
    ai += zv * wa[t];
    aj += zv * wa[HH + t];
  }
#pragma unroll
  for (int off = 16; off > 0; off >>= 1) {
    ai += __shfl_xor(ai, off, 32);
    aj += __shfl_xor(aj, off, 32);
  }
  if (lane == 0) {
    s_i[row] = ai;
    s_j[row] = aj;
  }
}

// ---------------------------------------------------------------------------
// K2b: per (b,i): stage dep[b,i,:,:] (128x64 = 32KB) in LDS once; compute
//   s_e, score = leaky_relu(s_i + s_j + s_e + ba), masked softmax -> attn,
//   D[b,i,e] = sum_j attn[j]*dep[j,e],  upd flag = asp_valid && any(adj row)
// Single HBM pass over the 134 MB dep tensor (roofline-dominant term).
// ---------------------------------------------------------------------------
__global__ __launch_bounds__(256) void attn_kernel(const float* __restrict__ dep,
                                                   const int* __restrict__ adj,
                                                   const float* __restrict__ wa,
                                                   const float* __restrict__ ba,
                                                   const float* __restrict__ s_i,
                                                   const float* __restrict__ s_j,
                                                   const int* __restrict__ asp_start,
                                                   const int* __restrict__ asp_end,
                                                   float* __restrict__ attn,
                                                   float* __restrict__ D,
                                                   int* __restrict__ upd) {
  __shared__ float sdep[LL * EE];   // 32 KB
  __shared__ float swae[EE];
  __shared__ float sattn[LL];
  __shared__ float red[LL];
  __shared__ float dred[256];
  __shared__ int anyNb;

  const int bi = blockIdx.x;        // 4096 blocks
  const int b = bi >> 7;
  const int i = bi & (LL - 1);
  const int tid = threadIdx.x;

  // cooperative 32KB slab load (float4)
  const float4* src = (const float4*)(dep + (size_t)bi * LL * EE);
  float4* dst = (float4*)sdep;
#pragma unroll
  for (int t = tid; t < (LL * EE) / 4; t += 256) dst[t] = src[t];
  if (tid < EE) swae[tid] = wa[2 * HH + tid];
  if (tid == 0) anyNb = 0;
  __syncthreads();

  const float si = s_i[bi];
  const float bav = ba[0];

  float val = -1e9f;
  float ex = 0.0f;
  int msk = 0;
  if (tid < LL) {
    msk = adj[(size_t)bi * LL + tid] > 0;
    if (msk) atomicOr(&anyNb, 1);
    const float* dj = sdep + tid * EE;
    float se = 0.0f;
#pragma unroll
    for (int e = 0; e < EE; ++e) se += dj[e] * swae[e];
    float sc = si + s_j[b * LL + tid] + se + bav;
    sc = sc > 0.0f ? sc : 0.01f * sc;           // leaky_relu(., 0.01)
    val = msk ? sc : -1e9f;
  }

  // max reduction over 128 masked scores
  if (tid < LL) red[tid] = val;
  __syncthreads();
  for (int s = 64; s > 0; s >>= 1) {
    if (tid < s) red[tid] = fmaxf(red[tid], red[tid + s]);
    __syncthreads();
  }
  const float mx = red[0];
  __syncthreads();

  if (tid < LL) ex = msk ? __expf(val - mx) : 0.0f;
  if (tid < LL) red[tid] = ex;
  __syncthreads();
  for (int s = 64; s > 0; s >>= 1) {
    if (tid < s) red[tid] = red[tid] + red[tid + s];
    __syncthreads();
  }
  const float sm = red[0];
  __syncthreads();

  const float at = (tid < LL && sm > 0.0f) ? ex / sm : 0.0f;
  if (tid < LL) {
    sattn[tid] = at;
    attn[(size_t)bi * LL + tid] = at;
  }
  __syncthreads();

  // D[b,i,e] = sum_j attn[j] * dep[j,e]   (256 threads = 64 e x 4 j-chunks)
  const int e = tid & (EE - 1);
  const int part = tid >> 6;
  float dsum = 0.0f;
#pragma unroll
  for (int j = part * 32; j < part * 32 + 32; ++j) dsum += sattn[j] * sdep[j * EE + e];
  dred[tid] = dsum;
  __syncthreads();
  if (tid < 128) dred[tid] += dred[tid + 128];
  __syncthreads();
  if (tid < 64) D[(size_t)bi * EE + tid] = dred[tid] + dred[tid + 64];

  if (tid == 0) {
    int valid = (i >= asp_start[b]) && (i <= asp_end[b]);
    upd[bi] = (valid && anyNb) ? 1 : 0;
  }
}

// ---------------------------------------------------------------------------
// K4: nr[b,i,h] = sum_j attn[b,i,j]*A[b,j,h] + sum_e D[b,i,e]*Wf[h, H+e]
// WMMA tiles: per batch, 16i x 16h, K over j (128) then e (64)
// ---------------------------------------------------------------------------
__global__ __launch_bounds__(256) void neighbor_kernel(const float* __restrict__ attn,
                                                       const float* __restrict__ Aw,
                                                       const float* __restrict__ D,
                                                       const float* __restrict__ Wf,
                                                       float* __restrict__ nr) {
  const int lane = threadIdx.x & 31;
  const int wave = threadIdx.x >> 5;
  const int tile = blockIdx.x * 8 + wave;       // 32*8*48 = 12288 tiles
  const int b = tile / (8 * NT);
  const int rem = tile % (8 * NT);
  const int it = rem / NT;
  const int ht = rem % NT;
  const int half = lane >> 4;
  const int col = lane & 15;
  const int i_lane = it * 16 + col;
  const int h_col = ht * 16 + col;
  const int koff = 2 * half;

  v8f acc = {};

  // attn @ A  (K = 128 over j)
  const float* arow = attn + (size_t)(b * LL + i_lane) * LL;
  const float* Ab = Aw + (size_t)b * LL * HH;
#pragma unroll 4
  for (int j0 = 0; j0 < LL; j0 += 4) {
    v2f a = *(const v2f*)(arow + j0 + koff);
    int k = j0 + koff;
    v2f bv;
    bv.x = Ab[(size_t)k * HH + h_col];
    bv.y = Ab[(size_t)(k + 1) * HH + h_col];
    acc = wmma4(a, bv, acc);
  }

  // D @ Wf[:, H:]^T  (K = 64 over e; Wf row stride = H+E = 832)
  const float* drow = D + (size_t)(b * LL + i_lane) * EE;
  const float* wrow = Wf + (size_t)h_col * (HH + EE) + HH;
#pragma unroll 4
  for (int e0 = 0; e0 < EE; e0 += 4) {
    v2f a = *(const v2f*)(drow + e0 + koff);
    v2f bv = *(const v2f*)(wrow + e0 + koff);
    acc = wmma4(a, bv, acc);
  }

#pragma unroll
  for (int r = 0; r < 8; ++r) {
    int m = b * LL + it * 16 + r + 8 * half;
    nr[(size_t)m * HH + h_col] = acc[r];
  }
}

// ---------------------------------------------------------------------------
// K5: temp = nr @ Wh[:, :H]^T + zs @ Wh[:, H:]^T, scatter row i -> out row
// (i+1)%L where upd[i] (== reference's roll(upd,1)/roll(temp,1) select)
// ---------------------------------------------------------------------------
__global__ __launch_bounds__(256) void final_kernel(const float* __restrict__ nr,
                                                    const float* __restrict__ zs,
                                                    const float* __restrict__ Wh,
                                                    const int* __restrict__ upd,
                                                    float* __restrict__ out) {
  const int lane = threadIdx.x & 31;
  const int wave = threadIdx.x >> 5;
  const int tile = blockIdx.x * 8 + wave;       // 256*48 = 12288 tiles
  const int mt = tile / NT;
  const int nt = tile % NT;
  const int half = lane >> 4;
  const int col = lane & 15;
  const int mrow = mt * 16 + col;
  const int n = nt * 16 + col;
  const int koff = 2 * half;

  const float* arow1 = nr + (size_t)mrow * HH;
  const float* arow2 = zs + (size_t)mrow * HH;
  const float* w1 = Wh + (size_t)n * (2 * HH);
  const float* w2 = w1 + HH;

  v8f acc = {};
#pragma unroll 4
  for (int k = 0; k < HH; k += 4) {
    v2f a = *(const v2f*)(arow1 + k + koff);
    v2f b = *(const v2f*)(w1 + k + koff);
    acc = wmma4(a, b, acc);
  }
#pragma unroll 4
  for (int k = 0; k < HH; k += 4) {
    v2f a = *(const v2f*)(arow2 + k + koff);
    v2f b = *(const v2f*)(w2 + k + koff);
    acc = wmma4(a, b, acc);
  }

#pragma unroll
  for (int r = 0; r < 8; ++r) {
    int m = mt * 16 + r + 8 * half;
    if (upd[m]) {
      int b = m >> 7, i = m & (LL - 1);
      int mdst = (b << 7) | ((i + 1) & (LL - 1));
      out[(size_t)mdst * HH + n] = acc[r];
    }
  }
}

// ---------------------------------------------------------------------------
// Host launcher
// ---------------------------------------------------------------------------
extern "C" void kernel_launch(void* const* d_in, const int* in_sizes, int n_in,
                              void* d_out, int out_size, void* d_ws, size_t ws_size,
                              hipStream_t stream) {
  const float* bert = (const float*)d_in[0];
  const float* dep  = (const float*)d_in[1];
  const int*   adj  = (const int*)d_in[2];
  const int*   aspS = (const int*)d_in[3];
  const int*   aspE = (const int*)d_in[4];
  const float* Wz   = (const float*)d_in[5];
  const float* bz   = (const float*)d_in[6];
  const float* wa   = (const float*)d_in[7];
  const float* ba   = (const float*)d_in[8];
  const float* Wf   = (const float*)d_in[9];
  const float* Wh   = (const float*)d_in[10];
  float* out = (float*)d_out;

  // workspace layout (floats)
  float* ws   = (float*)d_ws;
  float* zs   = ws;                       // 4096*768
  float* A    = zs + (size_t)MM * HH;     // 4096*768
  float* nr   = A + (size_t)MM * HH;      // 4096*768
  float* attn = nr + (size_t)MM * HH;     // 4096*128
  float* D    = attn + (size_t)MM * LL;   // 4096*64
  float* s_i  = D + (size_t)MM * EE;      // 4096
  float* s_j  = s_i + MM;                 // 4096
  int*   upd  = (int*)(s_j + MM);         // 4096

  // K0: default output = bert
  copy_f4_kernel<<<3072, 256, 0, stream>>>((const float4*)bert, (float4*)out,
                                           (MM * HH) / 4);
  // K1: zs = roll(bert @ Wz^T + bz, -1)
  gemm_wmma_kernel<<<1536, 256, 0, stream>>>(bert, /*roll*/1, Wz, HH, bz, zs, HH);
  // K2a: s_i, s_j
  score_ij_kernel<<<512, 256, 0, stream>>>(zs, wa, s_i, s_j);
  // K2b: attn, D, upd (single pass over dep)
  attn_kernel<<<MM, 256, 0, stream>>>(dep, adj, wa, ba, s_i, s_j, aspS, aspE,
                                      attn, D, upd);
  // K3: A = zs @ Wf[:, :H]^T  (row stride H+E)
  gemm_wmma_kernel<<<1536, 256, 0, stream>>>(zs, /*roll*/0, Wf, HH + EE, nullptr, A, HH);
  // K4: neighbor_repr
  neighbor_kernel<<<1536, 256, 0, stream>>>(attn, A, D, Wf, nr);
  // K5: temp + aspect-masked rolled scatter into out
  final_kernel<<<1536, 256, 0, stream>>>(nr, zs, Wh, upd, out);
}